// Sampler_32341103738936
// MI455X (gfx1250) — compile-verified
//
#include <hip/hip_runtime.h>
#include <hip/hip_bf16.h>
#include <stdint.h>
#include <stddef.h>

typedef float v4f __attribute__((ext_vector_type(4)));

#define BLOCK 1024        // 32 waves of 32 (wave32) -> one-shuffle cross-wave reduce
#define NW (BLOCK / 32)   // 32
#define STAGE_BYTES (BLOCK * (int)sizeof(v4f))  // 16 KB per double-buffer stage

__device__ __forceinline__ float waveReduceMax(float v) {
#pragma unroll
  for (int o = 16; o > 0; o >>= 1) v = fmaxf(v, __shfl_xor(v, o, 32));
  return v;
}

__device__ __forceinline__ float waveReduceSum(float v) {
#pragma unroll
  for (int o = 16; o > 0; o >>= 1) v += __shfl_xor(v, o, 32);
  return v;
}

// One workgroup per batch row. 3 phases over the row:
//   A: row max          (logits -> HBM once, then L2-resident: 512KB/row,
//                        256 rows * 512KB = 128MB < 192MB L2)
//   B: sum of exp(x-m)  (logits from L2)
//   C: write logits copy + probs (NT stores); q streamed once via
//      GLOBAL_LOAD_ASYNC_TO_LDS_B128 double-buffering (ASYNCcnt-tracked),
//      exponential-race argmax with first-index tie-break.
__global__ __launch_bounds__(BLOCK)
void sampler_softmax_race_async_kernel(const float* __restrict__ logits,
                                       const float* __restrict__ qu,
                                       float* __restrict__ outLogits,
                                       float* __restrict__ outProbs,
                                       float* __restrict__ outSampled,
                                       int V) {
  const int row  = blockIdx.x;
  const int tid  = threadIdx.x;
  const int lane = tid & 31;
  const int wv   = tid >> 5;
  const int nvec = V >> 2;  // V = 128000 -> 32000 float4 per row (16B aligned)

  const v4f* __restrict__ lrow = (const v4f*)(logits    + (size_t)row * V);
  const v4f* __restrict__ qrow = (const v4f*)(qu        + (size_t)row * V);
  v4f* __restrict__ o0         = (v4f*)(outLogits       + (size_t)row * V);
  v4f* __restrict__ o1         = (v4f*)(outProbs        + (size_t)row * V);

  __shared__ float sRed[NW];
  __shared__ float sMax;
  __shared__ float sSum;
  __shared__ float sRV[NW];
  __shared__ int   sRI[NW];
  __shared__ v4f   qstage[2][BLOCK];  // 32 KB async double buffer for q

  // ---------------- Phase A: row max ----------------
  float m = -INFINITY;
  for (int i = tid; i < nvec; i += BLOCK) {
    if (i + 2 * BLOCK < nvec) __builtin_prefetch(&lrow[i + 2 * BLOCK], 0, 3);
    v4f x = lrow[i];  // regular (RT) load: keep row hot in L2 for phases B/C
    m = fmaxf(fmaxf(fmaxf(m, x.x), fmaxf(x.y, x.z)), x.w);
  }
  m = waveReduceMax(m);
  if (lane == 0) sRed[wv] = m;
  __syncthreads();
  if (wv == 0) {
    float t = waveReduceMax(sRed[lane]);  // NW == 32: one wave covers all partials
    if (lane == 0) sMax = t;
  }
  __syncthreads();
  const float rowMax = sMax;

  // ---------------- Phase B: sum of exp(x - max) ----------------
  float s = 0.0f;
  for (int i = tid; i < nvec; i += BLOCK) {
    v4f x = lrow[i];  // L2 hit
    s += __expf(x.x - rowMax) + __expf(x.y - rowMax) +
         __expf(x.z - rowMax) + __expf(x.w - rowMax);
  }
  s = waveReduceSum(s);
  if (lane == 0) sRed[wv] = s;
  __syncthreads();
  if (wv == 0) {
    float t = waveReduceSum(sRed[lane]);
    if (lane == 0) sSum = t;
  }
  __syncthreads();
  const float invSum = 1.0f / sSum;

  // ---------------- Phase C: outputs + exponential-race argmax --------------
  // q tile k+1 is pulled into LDS by the async engine while tile k is consumed.
  // LDS address (ISA 15.18): the VDST VGPR holds the LDS byte offset; the low 32 bits of a
  // generic pointer to __shared__ are exactly that DS-style offset.
  const uint32_t ldsBase =
      (uint32_t)(uintptr_t)(&qstage[0][0]) + (uint32_t)tid * (uint32_t)sizeof(v4f);

  {
    const uint64_t g0 = (uint64_t)(uintptr_t)(qrow + tid);  // tile 0 (all lanes)
    asm volatile("global_load_async_to_lds_b128 %0, %1, off"
                 :: "v"(ldsBase), "v"(g0) : "memory");
  }

  float best = -1.0f;  // ratios are >= 0
  int   bidx = V;
  int   stage = 0;
  for (int i = tid; i < nvec; i += BLOCK) {
    // WAR guard: previous iteration's ds_load of the stage we are about to
    // overwrite must have completed before the async engine may write it.
    asm volatile("s_wait_dscnt 0x0" ::: "memory");

    const int inext = i + BLOCK;
    if (inext < nvec) {  // wave-uniform (boundary splits on 256-thread multiple)
      const uint32_t ldsNext = ldsBase + (uint32_t)((stage ^ 1) * STAGE_BYTES);
      const uint64_t gn = (uint64_t)(uintptr_t)(qrow + inext);
      asm volatile("global_load_async_to_lds_b128 %0, %1, off"
                   :: "v"(ldsNext), "v"(gn) : "memory");
      asm volatile("s_wait_asynccnt 0x1" ::: "memory");  // current tile landed
    } else {
      asm volatile("s_wait_asynccnt 0x0" ::: "memory");  // drain for last tile
    }

    v4f u = qstage[stage][tid];  // ds_load_b128 from the completed stage
    stage ^= 1;

    v4f x = lrow[i];  // L2 hit
    v4f p;
    p.x = __expf(x.x - rowMax) * invSum;
    p.y = __expf(x.y - rowMax) * invSum;
    p.z = __expf(x.z - rowMax) * invSum;
    p.w = __expf(x.w - rowMax) * invSum;

    __builtin_nontemporal_store(x, &o0[i]);  // logits32 copy, never re-read
    __builtin_nontemporal_store(p, &o1[i]);  // probs, never re-read

    // q = -log1p(-u); log1pf (not log(1-u)): u < 2^-25 happens ~once in 32.8M
    // samples and log(1-u)=0 would corrupt the argmax via a p/eps blow-up.
    const float r0 = p.x / (-log1pf(-u.x) + 1e-20f);
    const float r1 = p.y / (-log1pf(-u.y) + 1e-20f);
    const float r2 = p.z / (-log1pf(-u.z) + 1e-20f);
    const float r3 = p.w / (-log1pf(-u.w) + 1e-20f);

    const int base = i << 2;  // ascending order + strict '>' keeps first index
    if (r0 > best) { best = r0; bidx = base; }
    if (r1 > best) { best = r1; bidx = base + 1; }
    if (r2 > best) { best = r2; bidx = base + 2; }
    if (r3 > best) { best = r3; bidx = base + 3; }
  }

  // argmax reduce with first-index tie-break (wave32 shuffles)
#pragma unroll
  for (int o = 16; o > 0; o >>= 1) {
    float rv = __shfl_xor(best, o, 32);
    int   ri = __shfl_xor(bidx, o, 32);
    if (rv > best || (rv == best && ri < bidx)) { best = rv; bidx = ri; }
  }
  if (lane == 0) { sRV[wv] = best; sRI[wv] = bidx; }
  __syncthreads();
  if (wv == 0) {
    float rv = sRV[lane];
    int   ri = sRI[lane];
#pragma unroll
    for (int o = 16; o > 0; o >>= 1) {
      float rv2 = __shfl_xor(rv, o, 32);
      int   ri2 = __shfl_xor(ri, o, 32);
      if (rv2 > rv || (rv2 == rv && ri2 < ri)) { rv = rv2; ri = ri2; }
    }
    if (lane == 0) outSampled[row] = (float)ri;  // < 2^24: exact in f32
  }
}

extern "C" void kernel_launch(void* const* d_in, const int* in_sizes, int n_in,
                              void* d_out, int out_size, void* d_ws, size_t ws_size,
                              hipStream_t stream) {
  const float* logits = (const float*)d_in[0];
  const float* qu     = (const float*)d_in[1];

  const int V = 128000;
  const int B = in_sizes[0] / V;  // 256

  float* out        = (float*)d_out;
  float* outLogits  = out;                              // [B*V]
  float* outProbs   = out + (size_t)B * V;              // [B*V]
  float* outSampled = out + 2 * (size_t)B * V;          // [B]

  sampler_softmax_race_async_kernel<<<B, BLOCK, 0, stream>>>(
      logits, qu, outLogits, outProbs, outSampled, V);
}